// UserLevelAggregator_41369124995334
// MI455X (gfx1250) — compile-verified
//
#include <hip/hip_runtime.h>
#include <hip/hip_bf16.h>

typedef __attribute__((ext_vector_type(2))) float v2f;
typedef __attribute__((ext_vector_type(8))) float v8f;

#define U_USERS 16384
#define D_DIM   256
#define T_TRAIT 5
#define SPLIT   4              // D-dimension split: 4 waves per 16-user group
#define TILES   (16 / SPLIT)   // 4 dim-tiles (64 dims) per wave

// ---------------- Kernel 1: zero counts + ysum ----------------
__global__ void ula_zero_kernel(unsigned* __restrict__ counts, float* __restrict__ ysum) {
    int i = blockIdx.x * blockDim.x + threadIdx.x;
    if (i < U_USERS) counts[i] = 0u;
    if (i < U_USERS * T_TRAIT) ysum[i] = 0.0f;
}

// ---------------- Kernel 2: histogram + y atomic sums ----------------
__global__ void ula_hist_kernel(const int* __restrict__ user_ids,
                                const float* __restrict__ y,
                                unsigned* __restrict__ counts,
                                float* __restrict__ ysum, int n) {
    int i = blockIdx.x * blockDim.x + threadIdx.x;
    if (i >= n) return;
    int u = user_ids[i];
    atomicAdd(&counts[u], 1u);
#pragma unroll
    for (int t = 0; t < T_TRAIT; ++t)
        atomicAdd(&ysum[u * T_TRAIT + t], y[i * T_TRAIT + t]);
}

// ---------------- Kernel 3: exclusive prefix sum over 16384 counts (1 block) ----------------
__global__ __launch_bounds__(1024) void ula_scan_kernel(const unsigned* __restrict__ counts,
                                                        unsigned* __restrict__ offsets,
                                                        unsigned* __restrict__ cursor) {
    __shared__ unsigned tsum[1024];
    const int tid = threadIdx.x;
    unsigned local[16];
    unsigned s = 0;
#pragma unroll
    for (int k = 0; k < 16; ++k) { local[k] = counts[tid * 16 + k]; s += local[k]; }
    tsum[tid] = s;
    __syncthreads();
    for (int off = 1; off < 1024; off <<= 1) {
        unsigned v = (tid >= off) ? tsum[tid - off] : 0u;
        __syncthreads();
        tsum[tid] += v;
        __syncthreads();
    }
    unsigned run = tsum[tid] - s;
#pragma unroll
    for (int k = 0; k < 16; ++k) {
        offsets[tid * 16 + k] = run;
        cursor [tid * 16 + k] = run;
        run += local[k];
    }
    if (tid == 1023) offsets[U_USERS] = tsum[1023];
}

// ---------------- Kernel 4: scatter (user, post_index) pairs into sorted order ----------------
__global__ void ula_scatter_kernel(const int* __restrict__ user_ids,
                                   unsigned* __restrict__ cursor,
                                   uint2* __restrict__ sdata, int n) {
    int i = blockIdx.x * blockDim.x + threadIdx.x;
    if (i >= n) return;
    int u = user_ids[i];
    unsigned pos = atomicAdd(&cursor[u], 1u);
    uint2 v; v.x = (unsigned)u; v.y = (unsigned)i;
    sdata[pos] = v;   // one b64 store: {user, original row}
}

// ---------------- Kernel 5: WMMA segmented gather-reduce (D split 4 ways, 8 posts/iter) ----
// Wave w: group g = w>>2 (users g*16..g*16+15), split s = w&3 (dims s*64..s*64+63).
// V_WMMA_F32_16X16X4_F32 with segment-indicator A; scalar (readfirstlane) loop bounds
// keep EXEC all-ones through every WMMA; 2 chunks (8 posts) per iteration for MLP.
__global__ __launch_bounds__(256) void ula_wmma_reduce_kernel(
    const float* __restrict__ z,
    const uint2* __restrict__ sdata,
    const unsigned* __restrict__ offsets,
    const unsigned* __restrict__ counts,
    float* __restrict__ aggz) {
    const int wave = (blockIdx.x * (int)blockDim.x + (int)threadIdx.x) >> 5;
    const int lane = threadIdx.x & 31;
    const int u0     = (wave >> 2) * 16;
    const int dimoff = (wave & 3) * (TILES * 16);
    // Uniform loop bounds in SGPRs -> scalar loop, EXEC untouched.
    const unsigned start = (unsigned)__builtin_amdgcn_readfirstlane((int)offsets[u0]);
    const unsigned end   = (unsigned)__builtin_amdgcn_readfirstlane((int)offsets[u0 + 16]);
    const int col  = lane & 15;
    const unsigned kr = (lane < 16) ? 0u : 2u;
    const unsigned target = (unsigned)(u0 + col);

    v8f acc[TILES];
#pragma unroll
    for (int t = 0; t < TILES; ++t) acc[t] = (v8f){0,0,0,0,0,0,0,0};

    if (start < end) {
        const unsigned last = end - 1u;
        // Prologue: indices for the first two chunks (clamped; A forced 0 when OOB).
        uint2 c0 = sdata[(start + kr)       < last ? (start + kr)       : last];
        uint2 c1 = sdata[(start + kr + 1u)  < last ? (start + kr + 1u)  : last];
        uint2 d0 = sdata[(start + 4u + kr)      < last ? (start + 4u + kr)      : last];
        uint2 d1 = sdata[(start + 4u + kr + 1u) < last ? (start + 4u + kr + 1u) : last];
        for (unsigned j = start; j < end; j += 8u) {
            // Prefetch indices for the next two chunks (scalar select keeps loop uniform).
            unsigned nj = (j + 8u < end) ? (j + 8u) : j;
            unsigned q0 = nj + kr, q1 = q0 + 1u, q2 = nj + 4u + kr, q3 = q2 + 1u;
            uint2 n0 = sdata[q0 < last ? q0 : last];
            uint2 n1 = sdata[q1 < last ? q1 : last];
            uint2 m0 = sdata[q2 < last ? q2 : last];
            uint2 m1 = sdata[q3 < last ? q3 : last];

            // Indicator A columns for both chunks.
            float a00 = ((j + kr)          < end && c0.x == target) ? 1.0f : 0.0f;
            float a01 = ((j + kr + 1u)     < end && c1.x == target) ? 1.0f : 0.0f;
            float a10 = ((j + 4u + kr)     < end && d0.x == target) ? 1.0f : 0.0f;
            float a11 = ((j + 4u + kr + 1u)< end && d1.x == target) ? 1.0f : 0.0f;

            const float* z00 = z + (size_t)c0.y * D_DIM + dimoff + col;
            const float* z01 = z + (size_t)c1.y * D_DIM + dimoff + col;
            const float* z10 = z + (size_t)d0.y * D_DIM + dimoff + col;
            const float* z11 = z + (size_t)d1.y * D_DIM + dimoff + col;

            // Batch all 16 B loads before the WMMAs (deep load clause, one wait chain).
            float b00[TILES], b01[TILES], b10[TILES], b11[TILES];
#pragma unroll
            for (int t = 0; t < TILES; ++t) b00[t] = z00[t * 16];
#pragma unroll
            for (int t = 0; t < TILES; ++t) b01[t] = z01[t * 16];
#pragma unroll
            for (int t = 0; t < TILES; ++t) b10[t] = z10[t * 16];
#pragma unroll
            for (int t = 0; t < TILES; ++t) b11[t] = z11[t * 16];

            v2f A0; A0.x = a00; A0.y = a01;
            v2f A1; A1.x = a10; A1.y = a11;
#pragma unroll
            for (int t = 0; t < TILES; ++t) {
                v2f B; B.x = b00[t]; B.y = b01[t];
                acc[t] = __builtin_amdgcn_wmma_f32_16x16x4_f32(
                    false, A0, false, B, (short)0, acc[t], false, false);
            }
#pragma unroll
            for (int t = 0; t < TILES; ++t) {
                v2f B; B.x = b10[t]; B.y = b11[t];
                acc[t] = __builtin_amdgcn_wmma_f32_16x16x4_f32(
                    false, A1, false, B, (short)0, acc[t], false, false);
            }
            c0 = n0; c1 = n1; d0 = m0; d1 = m1;
        }
    }

    // Epilogue: divide by max(count,1), store 16 users x 64 dims.
#pragma unroll
    for (int r = 0; r < 8; ++r) {
        int m = (lane < 16) ? r : (r + 8);
        unsigned c = counts[u0 + m];
        float inv = 1.0f / (float)(c < 1u ? 1u : c);
        size_t rowbase = (size_t)(u0 + m) * D_DIM + dimoff + col;
#pragma unroll
        for (int t = 0; t < TILES; ++t)
            aggz[rowbase + (size_t)t * 16] = acc[t][r] * inv;
    }
}

// ---------------- Kernel 6: finalize y (mean + threshold) ----------------
__global__ void ula_finalize_y_kernel(const float* __restrict__ ysum,
                                      const unsigned* __restrict__ counts,
                                      float* __restrict__ out_y) {
    int i = blockIdx.x * blockDim.x + threadIdx.x;
    if (i >= U_USERS * T_TRAIT) return;
    int u = i / T_TRAIT;
    unsigned c = counts[u];
    float denom = (float)(c < 1u ? 1u : c);
    out_y[i] = (ysum[i] / denom >= 0.5f) ? 1.0f : 0.0f;
}

extern "C" void kernel_launch(void* const* d_in, const int* in_sizes, int n_in,
                              void* d_out, int out_size, void* d_ws, size_t ws_size,
                              hipStream_t stream) {
    const float* z        = (const float*)d_in[0];
    const int*   user_ids = (const int*)d_in[1];
    const float* y        = (const float*)d_in[2];
    const int N = in_sizes[1];

    float* out_z = (float*)d_out;                    // [U, D]
    float* out_y = out_z + (size_t)U_USERS * D_DIM;  // [U, T]

    // Workspace layout
    char* ws = (char*)d_ws;
    size_t o = 0;
    unsigned* counts  = (unsigned*)(ws + o); o += (size_t)U_USERS * 4;
    unsigned* offsets = (unsigned*)(ws + o); o += (size_t)(U_USERS + 4) * 4;
    unsigned* cursor  = (unsigned*)(ws + o); o += (size_t)U_USERS * 4;
    float*    ysum    = (float*)   (ws + o); o += (size_t)U_USERS * T_TRAIT * 4;
    o = (o + 15) & ~(size_t)15;
    uint2*    sdata   = (uint2*)   (ws + o); o += (size_t)N * 8;
    (void)ws_size; (void)n_in; (void)out_size;

    const int B = 256;
    ula_zero_kernel<<<(U_USERS * T_TRAIT + B - 1) / B, B, 0, stream>>>(counts, ysum);
    ula_hist_kernel<<<(N + B - 1) / B, B, 0, stream>>>(user_ids, y, counts, ysum, N);
    ula_scan_kernel<<<1, 1024, 0, stream>>>(counts, offsets, cursor);
    ula_scatter_kernel<<<(N + B - 1) / B, B, 0, stream>>>(user_ids, cursor, sdata, N);
    // (U/16) groups * SPLIT waves = 4096 waves; 8 waves per 256-thread block -> 512 blocks.
    ula_wmma_reduce_kernel<<<(U_USERS / 16) * SPLIT * 32 / B, B, 0, stream>>>(
        z, sdata, offsets, counts, out_z);
    ula_finalize_y_kernel<<<(U_USERS * T_TRAIT + B - 1) / B, B, 0, stream>>>(ysum, counts, out_y);
}